// MyModel_7181185319678
// MI455X (gfx1250) — compile-verified
//
#include <hip/hip_runtime.h>

// ---------------------------------------------------------------------------
// Vanilla tanh-RNN (H=16, in=1, out=1), B=4096 sequences x T=1024 steps.
//
// Per wave (wave32), per 16-batch tile, computed transposed:
//   H'[j][b] = tanh( x[b][t]*W_ih[j] + b_ih[j]+b_hh[j] + sum_k W_hh[j][k]*H[k][b] )
// as D = A x B + C with V_WMMA_F32_16X16X4_F32 chained 4x (K=16):
//   A = W_hh      (loop invariant, A-layout)
//   B = H         (hidden-major x batch, B-layout)
//   C = x_proj^T  (C/D layout)
// D -> next-step B operand is a pure half-wave exchange: one
// v_permlane16_swap_b32 per register pair (VALU, no LDS pipe), falling back
// to shfl_xor+select if the builtin is unavailable.
// tanh is a single hw transcendental (v_tanh_f32) -> branch-free inner loop.
// Linear head: 8 FMAs, then one permlane-swap + add produces the fully
// reduced y for BOTH chains in store position (lanes 0-15 chain0,
// lanes 16-31 chain1); one b128 y store per 4 steps.
//
// 2 independent 16-batch chains per wave => 2-way ILP across the serial
// WMMA->tanh->WMMA dependency chain. Grid: 4096/32 = 128 waves.
// ---------------------------------------------------------------------------

namespace {
constexpr int kB = 4096;
constexpr int kT = 1024;
constexpr int kH = 16;
}

typedef __attribute__((ext_vector_type(2))) float v2f;
typedef __attribute__((ext_vector_type(4))) float v4f;
typedef __attribute__((ext_vector_type(8))) float v8f;
typedef __attribute__((ext_vector_type(2))) int   v2i;

__device__ __forceinline__ float xor16(float v) {
    return __shfl_xor(v, 16, 32);   // swap lane halves (wave32)
}

__device__ __forceinline__ float fast_tanh(float v) {
#if __has_builtin(__builtin_amdgcn_tanhf)
    return __builtin_amdgcn_tanhf(v);           // v_tanh_f32 (CDNA5 trans op)
#else
    float e = __builtin_amdgcn_exp2f(v * 2.88539008177792681472f);
    return 1.0f - 2.0f * __builtin_amdgcn_rcpf(e + 1.0f);
#endif
}

// Half-wave exchange of two registers:
//   o1 = {a[0:15],  b[0:15]}   (b's low-half values placed in lanes 16-31)
//   o2 = {a[16:31], b[16:31]}  (a's high-half values placed in lanes 0-15)
__device__ __forceinline__ void half_swap(int hi, float a, float b,
                                          float& o1, float& o2) {
#if __has_builtin(__builtin_amdgcn_permlane16_swap)
    (void)hi;
    v2i r = __builtin_amdgcn_permlane16_swap(__float_as_int(a),
                                             __float_as_int(b),
                                             false, false);
    o1 = __int_as_float(r.x);
    o2 = __int_as_float(r.y);
#else
    o1 = hi ? xor16(b) : a;
    o2 = hi ? b : xor16(a);
#endif
}

__global__ __launch_bounds__(32) void rnn16_wmma_f32(
    const float* __restrict__ x,      // [B, T]
    const float* __restrict__ h0,     // [B, 16]
    const float* __restrict__ W_ih,   // [16]
    const float* __restrict__ W_hh,   // [16, 16]
    const float* __restrict__ b_ih,   // [16]
    const float* __restrict__ b_hh,   // [16]
    const float* __restrict__ W_lin,  // [16]
    const float* __restrict__ b_lin,  // [1]
    float* __restrict__ y_out,        // [B, T]
    float* __restrict__ h_out)        // [B, 16]
{
    const int lane = threadIdx.x;      // 0..31, one wave per workgroup
    const int lm   = lane & 15;        // batch-in-tile / M row
    const int hi   = lane >> 4;        // 0: lanes 0-15, 1: lanes 16-31
    int b0[2];
    b0[0] = (int)blockIdx.x * 32;
    b0[1] = b0[0] + 16;

    // ---- loop-invariant operands (shared by both chains) ----
    // A = W_hh in A-layout: chunk c, vgpr r, lane L -> W_hh[L%16][4c + r + 2*hi]
    v2f Amat[4];
#pragma unroll
    for (int c = 0; c < 4; ++c) {
        Amat[c].x = W_hh[lm * kH + 4 * c + 0 + 2 * hi];
        Amat[c].y = W_hh[lm * kH + 4 * c + 1 + 2 * hi];
    }
    // Per-lane slices for C operand and linear head: j = v + 8*hi
    float w8[8], bias8[8], wl8[8];
#pragma unroll
    for (int v = 0; v < 8; ++v) {
        const int j = v + 8 * hi;
        w8[v]    = W_ih[j];
        bias8[v] = b_ih[j] + b_hh[j];
        wl8[v]   = W_lin[j];
    }
    const float blin = b_lin[0];

    // ---- recurrent state: H in B-layout, Bst[s][2c+r] = H[4c+r+2*hi][lm] ----
    float Bst[2][8];
#pragma unroll
    for (int s = 0; s < 2; ++s)
#pragma unroll
        for (int c = 0; c < 4; ++c)
#pragma unroll
            for (int r = 0; r < 2; ++r)
                Bst[s][2 * c + r] = h0[(b0[s] + lm) * kH + 4 * c + r + 2 * hi];

    const float* xp[2];
#pragma unroll
    for (int s = 0; s < 2; ++s)
        xp[s] = x + (size_t)(b0[s] + lm) * kT;          // one x scalar/lane/step
    // full-wave y store: lane L covers batch b0[0]+L (chain = hi)
    float* yst = y_out + (size_t)(b0[0] + lane) * kT;

    for (int t0 = 0; t0 < kT; t0 += 4) {
        // stream x: one float4 per lane per chain per 4 steps
        v4f xv[2];
#pragma unroll
        for (int s = 0; s < 2; ++s)
            xv[s] = *(const v4f*)(xp[s] + t0);
        if (t0 + 64 < kT) {
            __builtin_prefetch(xp[0] + t0 + 64, 0, 1);
            __builtin_prefetch(xp[1] + t0 + 64, 0, 1);
        }

        v4f yv;                        // 4 steps of y for this lane's batch
#pragma unroll
        for (int u = 0; u < 4; ++u) {
            float p2[2];
#pragma unroll
            for (int s = 0; s < 2; ++s) {   // 2 independent chains -> ILP
                const float xs = xv[s][u];
                // C operand: x_proj^T tile
                v8f acc;
#pragma unroll
                for (int v = 0; v < 8; ++v)
                    acc[v] = xs * w8[v] + bias8[v];
                // 4 chained f32 WMMAs: full K=16 matvec for 16 batches
#pragma unroll
                for (int c = 0; c < 4; ++c) {
                    v2f bop;
                    bop.x = Bst[s][2 * c + 0];
                    bop.y = Bst[s][2 * c + 1];
                    acc = __builtin_amdgcn_wmma_f32_16x16x4_f32(
                        false, Amat[c], false, bop, (short)0, acc, false, false);
                }
                // activation: single hw transcendental per element
                float h[8];
#pragma unroll
                for (int v = 0; v < 8; ++v) h[v] = fast_tanh(acc[v]);
                // D(C/D layout) -> B layout: 4 half-wave register exchanges
                //   pair (h[a], h[a+2]) -> (B[k_lo], B[k_lo+8])
                half_swap(hi, h[0], h[2], Bst[s][0], Bst[s][4]); // K 0/2 | 8/10
                half_swap(hi, h[1], h[3], Bst[s][1], Bst[s][5]); // K 1/3 | 9/11
                half_swap(hi, h[4], h[6], Bst[s][2], Bst[s][6]); // K 4/6 | 12/14
                half_swap(hi, h[5], h[7], Bst[s][3], Bst[s][7]); // K 5/7 | 13/15
                // linear head partial (j-range of this half-wave)
                float p = 0.f;
#pragma unroll
                for (int v = 0; v < 8; ++v) p += h[v] * wl8[v];
                p2[s] = p;
            }
            // One half-swap combines BOTH chains' partials into store position:
            //   q1 = {pa.lo, pb.lo}, q2 = {pa.hi, pb.hi};  q1+q2 = full dot
            //   lanes 0-15 -> chain0's y, lanes 16-31 -> chain1's y
            float q1, q2;
            half_swap(hi, p2[0], p2[1], q1, q2);
            yv[u] = q1 + q2 + blin;
        }
        *(v4f*)(yst + t0) = yv;                  // one b128 store / 4 steps
    }

    // h_last: Bst holds final H in B-layout -> scatter back to [B,16]
#pragma unroll
    for (int s = 0; s < 2; ++s)
#pragma unroll
        for (int c = 0; c < 4; ++c)
#pragma unroll
            for (int r = 0; r < 2; ++r)
                h_out[(b0[s] + lm) * kH + 4 * c + r + 2 * hi] =
                    Bst[s][2 * c + r];
}

extern "C" void kernel_launch(void* const* d_in, const int* in_sizes, int n_in,
                              void* d_out, int out_size, void* d_ws, size_t ws_size,
                              hipStream_t stream) {
    (void)in_sizes; (void)n_in; (void)d_ws; (void)ws_size; (void)out_size;
    const float* x     = (const float*)d_in[0];   // [4096,1024,1]
    const float* h0    = (const float*)d_in[1];   // [1,4096,16]
    const float* W_ih  = (const float*)d_in[2];   // [16,1]
    const float* W_hh  = (const float*)d_in[3];   // [16,16]
    const float* b_ih  = (const float*)d_in[4];   // [16]
    const float* b_hh  = (const float*)d_in[5];   // [16]
    const float* W_lin = (const float*)d_in[6];   // [1,16]
    const float* b_lin = (const float*)d_in[7];   // [1]

    float* y_out = (float*)d_out;                       // [1, B*T, 1] flat
    float* h_out = y_out + (size_t)kB * kT;             // [1, B, 16] flat

    dim3 grid(kB / 32), block(32);                      // 1 wave/WG, 2 chains/wave
    rnn16_wmma_f32<<<grid, block, 0, stream>>>(
        x, h0, W_ih, W_hh, b_ih, b_hh, W_lin, b_lin, y_out, h_out);
}